// GAT_62182536511740
// MI455X (gfx1250) — compile-verified
//
#include <hip/hip_runtime.h>
#include <math.h>

// GAT-2-layer for MI455X (gfx1250). GEMMs use V_WMMA_F32_16X16X4_F32 (fp32
// precision preserved; workload is scatter/gather bound so fp32 WMMA costs
// nothing vs bf16). Edge softmax/aggregation via coalesced wave-per-(edge,head)
// atomics.

#define NHEAD 4
#define DTOT 128           // heads * per-head dim (both layers)
#define NEG_SLOPE 0.2f

typedef float v2f __attribute__((ext_vector_type(2)));
typedef float v8f __attribute__((ext_vector_type(8)));

// ---------------------------------------------------------------- utilities
__global__ void fill_kernel(float* __restrict__ p, int n, float v) {
  int tid = blockIdx.x * blockDim.x + threadIdx.x;
  if (tid < n) p[tid] = v;
}

__device__ __forceinline__ void atomicMaxFloat(float* addr, float value) {
  // sign-magnitude ordering trick: works for mixed signs when initialized -inf
  if (value >= 0.0f) atomicMax((int*)addr, __float_as_int(value));
  else               atomicMin((unsigned int*)addr, __float_as_uint(value));
}

// Wt[n][k] = W[k][n]  (128x128, one-time 64KB transpose so B fragments are
// contiguous b64 loads in the WMMA loop)
__global__ void transpose128_kernel(const float* __restrict__ W,
                                    float* __restrict__ Wt) {
  int tid = blockIdx.x * blockDim.x + threadIdx.x;   // 16384 threads
  if (tid >= DTOT * DTOT) return;
  int k = tid >> 7, n = tid & (DTOT - 1);
  Wt[(size_t)n * DTOT + k] = W[tid];
}

// ---------------------------------------------------------------- GEMM
// C[nRows x 128] = X[nRows x 128] @ W[128 x 128], fp32 WMMA 16x16x4.
// One block = 16-row strip; 8 waves cover the 8 column tiles.
// Wt is the transposed weight (Wt[n][k]).
__global__ void gemm128_wmma_kernel(const float* __restrict__ X,
                                    const float* __restrict__ Wt,
                                    float* __restrict__ C, int nRows) {
  const int lane = threadIdx.x & 31;
  const int wv   = threadIdx.x >> 5;     // 0..7 -> column tile
  const int r    = lane & 15;
  const int hf   = lane >> 4;            // half-wave: K offset selector
  const int rowBase = blockIdx.x << 4;
  const int colBase = wv << 4;

  int row = rowBase + r;
  if (row >= nRows) row = nRows - 1;     // clamp (EXEC must stay all-ones for WMMA)
  const float* __restrict__ Xp = X + (size_t)row * DTOT;
  const float* __restrict__ Wp = Wt + (size_t)(colBase + r) * DTOT;

  v8f acc = {};
  #pragma unroll 4
  for (int k0 = 0; k0 < DTOT; k0 += 4) {
    const int ka = k0 + (hf << 1);
    v2f a, b;
    // A 16x4 fp32 layout: lanes 0-15 K={k0,k0+1}, lanes 16-31 K={k0+2,k0+3}
    a.x = Xp[ka];
    a.y = Xp[ka + 1];
    // B 4x16 fp32 layout mirrors A over K; lanes index N. Wt row -> contiguous.
    b.x = Wp[ka];
    b.y = Wp[ka + 1];
    acc = __builtin_amdgcn_wmma_f32_16x16x4_f32(false, a, false, b,
                                                (short)0, acc, false, false);
  }
  // C/D layout: VGPR i -> row i (lanes 0-15) / row i+8 (lanes 16-31)
  const int orow = rowBase + (hf << 3);
  if (rowBase + 16 <= nRows) {
    // fast path: full tile, unguarded stores
    float* __restrict__ Cp = C + (size_t)orow * DTOT + colBase + r;
    #pragma unroll
    for (int i = 0; i < 8; ++i) Cp[i * DTOT] = acc[i];
  } else {
    #pragma unroll
    for (int i = 0; i < 8; ++i)
      if (orow + i < nRows)
        C[(size_t)(orow + i) * DTOT + colBase + r] = acc[i];
  }
}

// ---------------------------------------------------------------- scores
// es[n,h] = dot(h[n,h,:], a_src[h,:]);  ed likewise.
__global__ void scores_kernel(const float* __restrict__ h,
                              const float* __restrict__ as,
                              const float* __restrict__ ad,
                              float* __restrict__ es, float* __restrict__ ed,
                              int n) {
  int tid = blockIdx.x * blockDim.x + threadIdx.x;
  if (tid >= n * NHEAD) return;
  int node = tid >> 2, head = tid & 3;
  const float* hp = h + (size_t)node * DTOT + head * 32;
  const float* ap = as + head * 32;
  const float* bp = ad + head * 32;
  float s = 0.f, t = 0.f;
  #pragma unroll
  for (int d = 0; d < 32; ++d) {
    float v = hp[d];
    s += v * ap[d];
    t += v * bp[d];
  }
  es[tid] = s;
  ed[tid] = t;
}

// ---------------------------------------------------------------- edge pass 1
// e = leaky_relu(es[src]+ed[dst]); segment max into m[dst].
__global__ void edge_max_kernel(const int* __restrict__ src,
                                const int* __restrict__ dst,
                                const float* __restrict__ es,
                                const float* __restrict__ ed,
                                float* __restrict__ e_buf,
                                float* __restrict__ m, int E_) {
  int tid = blockIdx.x * blockDim.x + threadIdx.x;
  if (tid >= E_ * NHEAD) return;
  int e = tid >> 2, h = tid & 3;
  int s = src[e], d = dst[e];
  float v = es[s * NHEAD + h] + ed[d * NHEAD + h];
  v = v > 0.f ? v : NEG_SLOPE * v;
  e_buf[tid] = v;
  atomicMaxFloat(&m[d * NHEAD + h], v);
}

// ---------------------------------------------------------------- edge pass 2
// w = exp(e - m[dst]); den[dst] += w.
__global__ void edge_exp_kernel(const int* __restrict__ dst,
                                float* __restrict__ e_buf,
                                const float* __restrict__ m,
                                float* __restrict__ den, int E_) {
  int tid = blockIdx.x * blockDim.x + threadIdx.x;
  if (tid >= E_ * NHEAD) return;
  int e = tid >> 2, h = tid & 3;
  int d = dst[e];
  float w = expf(e_buf[tid] - m[d * NHEAD + h]);
  e_buf[tid] = w;
  atomicAdd(&den[d * NHEAD + h], w);
}

// ---------------------------------------------------------------- edge pass 2.5
// alpha = w / den[dst]  (keeps the 32x-replicated agg kernel divide-free)
__global__ void edge_alpha_kernel(const int* __restrict__ dst,
                                  float* __restrict__ e_buf,
                                  const float* __restrict__ den, int E_) {
  int tid = blockIdx.x * blockDim.x + threadIdx.x;
  if (tid >= E_ * NHEAD) return;
  int e = tid >> 2, h = tid & 3;
  e_buf[tid] = e_buf[tid] / den[dst[e] * NHEAD + h];
}

// ---------------------------------------------------------------- edge pass 3
// wave32 per (edge, head); lane = feature dim. Coalesced gather + atomic scatter.
__global__ void edge_agg_kernel(const int* __restrict__ src,
                                const int* __restrict__ dst,
                                const float* __restrict__ e_buf,
                                const float* __restrict__ h,
                                float* __restrict__ acc, int E_) {
  int gtid = blockIdx.x * blockDim.x + threadIdx.x;
  int wid  = gtid >> 5;
  int lane = gtid & 31;
  if (wid >= E_ * NHEAD) return;
  int e = wid >> 2, hd = wid & 3;
  int s = src[e], d = dst[e];
  float alpha = e_buf[wid];
  float v = alpha * h[(size_t)s * DTOT + hd * 32 + lane];
  atomicAdd(&acc[(size_t)d * DTOT + hd * 32 + lane], v);
}

// ---------------------------------------------------------------- bias + relu
__global__ void bias_relu_kernel(float* __restrict__ acc,
                                 const float* __restrict__ b, int total) {
  int tid = blockIdx.x * blockDim.x + threadIdx.x;
  if (tid >= total) return;
  float v = acc[tid] + b[tid & (DTOT - 1)];
  acc[tid] = v > 0.f ? v : 0.f;
}

// ---------------------------------------------------------------- log_softmax
// wave32 per row of 128; adds bias b2 first (reference adds bias, then lsm).
__global__ void log_softmax_kernel(float* __restrict__ out,
                                   const float* __restrict__ b, int n) {
  int gtid = blockIdx.x * blockDim.x + threadIdx.x;
  int row  = gtid >> 5;
  int lane = gtid & 31;
  if (row >= n) return;
  float* p = out + (size_t)row * DTOT;
  float v[4];
  float mx = -INFINITY;
  #pragma unroll
  for (int i = 0; i < 4; ++i) {
    v[i] = p[lane + 32 * i] + b[lane + 32 * i];
    mx = fmaxf(mx, v[i]);
  }
  #pragma unroll
  for (int off = 16; off > 0; off >>= 1) mx = fmaxf(mx, __shfl_xor(mx, off, 32));
  float sum = 0.f;
  #pragma unroll
  for (int i = 0; i < 4; ++i) sum += expf(v[i] - mx);
  #pragma unroll
  for (int off = 16; off > 0; off >>= 1) sum += __shfl_xor(sum, off, 32);
  float lse = logf(sum) + mx;
  #pragma unroll
  for (int i = 0; i < 4; ++i) p[lane + 32 * i] = v[i] - lse;
}

// ---------------------------------------------------------------- host side
static void run_gat_layer(const float* Xin, const float* W, const float* asrc,
                          const float* adst, const int* src, const int* dstv,
                          float* hbuf, float* acc, float* e_buf, float* es,
                          float* ed, float* m, float* den, float* Wt,
                          int N_, int E_, hipStream_t stream) {
  const int T = 256;
  // transpose weights (64KB, one-time per layer)
  transpose128_kernel<<<(DTOT * DTOT + T - 1) / T, T, 0, stream>>>(W, Wt);
  // h = Xin @ W  (WMMA fp32)
  gemm128_wmma_kernel<<<(N_ + 15) / 16, 256, 0, stream>>>(Xin, Wt, hbuf, N_);
  // per-node attention scores
  int nh = N_ * NHEAD;
  scores_kernel<<<(nh + T - 1) / T, T, 0, stream>>>(hbuf, asrc, adst, es, ed, N_);
  // init reductions + accumulator
  fill_kernel<<<(nh + T - 1) / T, T, 0, stream>>>(m, nh, -INFINITY);
  fill_kernel<<<(nh + T - 1) / T, T, 0, stream>>>(den, nh, 0.f);
  int tot = N_ * DTOT;
  fill_kernel<<<(tot + T - 1) / T, T, 0, stream>>>(acc, tot, 0.f);
  // edge softmax + aggregation
  int eh = E_ * NHEAD;
  edge_max_kernel<<<(eh + T - 1) / T, T, 0, stream>>>(src, dstv, es, ed, e_buf, m, E_);
  edge_exp_kernel<<<(eh + T - 1) / T, T, 0, stream>>>(dstv, e_buf, m, den, E_);
  edge_alpha_kernel<<<(eh + T - 1) / T, T, 0, stream>>>(dstv, e_buf, den, E_);
  long long aggThreads = (long long)eh * 32;
  unsigned aggBlocks = (unsigned)((aggThreads + T - 1) / T);
  edge_agg_kernel<<<aggBlocks, T, 0, stream>>>(src, dstv, e_buf, hbuf, acc, E_);
}

extern "C" void kernel_launch(void* const* d_in, const int* in_sizes, int n_in,
                              void* d_out, int out_size, void* d_ws, size_t ws_size,
                              hipStream_t stream) {
  (void)n_in; (void)out_size; (void)ws_size;
  const int*   edge = (const int*)d_in[0];
  const float* X    = (const float*)d_in[1];
  const float* W1   = (const float*)d_in[2];
  const float* a1s  = (const float*)d_in[3];
  const float* a1d  = (const float*)d_in[4];
  const float* b1   = (const float*)d_in[5];
  const float* W2   = (const float*)d_in[6];
  const float* a2s  = (const float*)d_in[7];
  const float* a2d  = (const float*)d_in[8];
  const float* b2   = (const float*)d_in[9];
  float* out = (float*)d_out;

  const int E_ = in_sizes[0] / 2;
  const int N_ = in_sizes[1] / DTOT;
  const int* src  = edge;
  const int* dstv = edge + E_;

  // workspace layout (floats)
  float* ws    = (float*)d_ws;
  float* hbuf  = ws;                                  // N*128 (h1, then h2)
  float* x1    = hbuf  + (size_t)N_ * DTOT;           // N*128 (layer1 output)
  float* e_buf = x1    + (size_t)N_ * DTOT;           // E*4
  float* es    = e_buf + (size_t)E_ * NHEAD;          // N*4
  float* ed    = es    + (size_t)N_ * NHEAD;          // N*4
  float* m     = ed    + (size_t)N_ * NHEAD;          // N*4
  float* den   = m     + (size_t)N_ * NHEAD;          // N*4
  float* Wt    = den   + (size_t)N_ * NHEAD;          // 128*128 transposed W

  const int T = 256;

  // ----- layer 1: x1 = relu(GAT(X) + b1)
  run_gat_layer(X, W1, a1s, a1d, src, dstv, hbuf, x1, e_buf, es, ed, m, den, Wt,
                N_, E_, stream);
  int tot = N_ * DTOT;
  bias_relu_kernel<<<(tot + T - 1) / T, T, 0, stream>>>(x1, b1, tot);

  // ----- layer 2: out = log_softmax(GAT(x1) + b2)
  run_gat_layer(x1, W2, a2s, a2d, src, dstv, hbuf, out, e_buf, es, ed, m, den, Wt,
                N_, E_, stream);
  long long lsmThreads = (long long)N_ * 32;
  unsigned lsmBlocks = (unsigned)((lsmThreads + T - 1) / T);
  log_softmax_kernel<<<lsmBlocks, T, 0, stream>>>(out, b2, N_);
}